// ManualAdaptiveTransition_48610439856350
// MI455X (gfx1250) — compile-verified
//
#include <hip/hip_runtime.h>

typedef __attribute__((ext_vector_type(2))) float v2f;
typedef __attribute__((ext_vector_type(4))) float v4f;
typedef __attribute__((ext_vector_type(8))) float v8f;

static constexpr int SN    = 128;        // state dim
static constexpr int TILE  = 16;         // WMMA tile
static constexpr int NB    = SN / TILE;  // 8 blocks
static constexpr int APAD  = 132;        // LDS row stride for A (conflict-free b64 frag reads)
static constexpr int XPAD  = 20;         // LDS row stride for 16x16 staging tiles
static constexpr int WAVES = 2;          // 64-thread WGs -> 128 WGs, ~1 wave/SIMD:
                                         // all serial diagonal solves run in parallel

// Bilinear (Tustin) HiPPO-LegS update:
//   x = (I - dt/2 A)^{-1} [ (I + dt/2 A) u + dt B v ]
// Single-accumulator formulation: at block-row k,
//   rhs_k = u_k + dt2*(A@u)_k + (dt*v)*B_k + sum_{j<k} A(k,j) @ Z_j ,  Z_j = dt2*y_j
// so only one v8f accumulator is live (no spills); solved Z blocks live in LDS.
__global__ __launch_bounds__(WAVES * 32) void hippo_bilinear_wmma(
    const float* __restrict__ u, const float* __restrict__ v,
    const float* __restrict__ delta, const float* __restrict__ A,
    const float* __restrict__ Bvec, float* __restrict__ out, int batch)
{
    __shared__ float A_lds[SN * APAD];                 // 67,584 B, shared by the WG
    __shared__ float xt[WAVES][TILE][XPAD];            // per-wave diagonal-solve staging
    __shared__ float zf[WAVES][NB][TILE][XPAD];        // per-wave Z_j = dt2*y_j, transposed [col][k]

    const int tid = threadIdx.x;
    const int nthreads = WAVES * 32;

    // ---- Phase 0: cooperatively load A (64 KB) into LDS via b128, padded rows ----
    #pragma unroll 4
    for (int i = 0; i < (SN * SN / 4) / nthreads; ++i) {
        int q = tid + i * nthreads;                    // float4 index
        int r = q >> 5;                                // /32 float4s per row
        int c = (q & 31) << 2;
        v4f t = *(const v4f*)(A + r * SN + c);
        *(v4f*)(&A_lds[r * APAD + c]) = t;
    }
    __syncthreads();

    const int wave = tid >> 5;
    const int lane = tid & 31;
    const int col  = lane & 15;      // batch column within tile / M-row of A frag
    const int hlf  = lane >> 4;      // half-wave selects K pair in f32 fragments
    const int batch_base = (blockIdx.x * WAVES + wave) * TILE;
    const int b = batch_base + col;
    if (batch_base >= batch) return;

    const float dt2 = 0.5f * delta[b];
    const float dvb = delta[b] * v[b];
    const float* ucol = u + (size_t)b * SN;            // this lane's batch column

    float* myxt = &xt[wave][0][0];

    // ---- Fused forward matvec + blocked forward substitution, one block-row at a time ----
    #pragma unroll 1
    for (int k = 0; k < NB; ++k) {
        const float* arows = &A_lds[(k * TILE + col) * APAD];

        // (A @ u)_k : sum_{j<=k} A(k,j) @ U_j
        v8f acc = {};
        for (int j = 0; j <= k; ++j) {
            const float* arow = arows + j * TILE;
            const float* ucb  = ucol + j * TILE;
            #pragma unroll
            for (int k4 = 0; k4 < 4; ++k4) {
                int ko = k4 * 4 + 2 * hlf;                       // K pair for this half
                v2f af = *(const v2f*)(arow + ko);               // A[M=col][K pair], LDS b64
                v2f bf = *(const v2f*)(ucb + ko);                // U[K][N=col], global b64
                acc = __builtin_amdgcn_wmma_f32_16x16x4_f32(
                          false, af, false, bf, (short)0, acc, false, false);
            }
        }

        // rhs_k = u_k + dt2*acc + (dt*v)*B_k   (rows M = r + 8*hlf, column = col)
        {
            const int nbase = k * TILE + 8 * hlf;
            const v4f u0 = *(const v4f*)(ucol + nbase);
            const v4f u1 = *(const v4f*)(ucol + nbase + 4);
            const v4f b0 = *(const v4f*)(Bvec + nbase);
            const v4f b1 = *(const v4f*)(Bvec + nbase + 4);
            #pragma unroll
            for (int r = 0; r < 4; ++r) {
                acc[r]     = u0[r] + dt2 * acc[r]     + dvb * b0[r];
                acc[r + 4] = u1[r] + dt2 * acc[r + 4] + dvb * b1[r];
            }
        }

        // trailing substitution contributions: sum_{j<k} A(k,j) @ Z_j  (all from LDS)
        for (int j = 0; j < k; ++j) {
            const float* arow = arows + j * TILE;
            const float* zrow = &zf[wave][j][col][0];            // transposed Z_j [col][kk]
            #pragma unroll
            for (int k4 = 0; k4 < 4; ++k4) {
                int ko = k4 * 4 + 2 * hlf;
                v2f af = *(const v2f*)(arow + ko);
                v2f zb = *(const v2f*)(zrow + ko);
                acc = __builtin_amdgcn_wmma_f32_16x16x4_f32(
                          false, af, false, zb, (short)0, acc, false, false);
            }
        }

        // stage rhs block k into LDS (C/D layout -> row-major tile)
        #pragma unroll
        for (int r = 0; r < 8; ++r)
            myxt[(r + 8 * hlf) * XPAD + col] = acc[r];
        asm volatile("s_wait_dscnt 0x0" ::: "memory");

        // serial 16x16 lower-triangular solve: one batch column per lane (0..15)
        if (lane < 16) {
            const int rb = k * TILE;
            float* myz = &zf[wave][k][lane][0];
            float y[16];
            #pragma unroll
            for (int m = 0; m < 16; ++m) {
                float s = myxt[m * XPAD + lane];
                float accv = 0.f;
                #pragma unroll
                for (int j = 0; j < m; ++j)
                    accv += A_lds[(rb + m) * APAD + rb + j] * y[j];   // LDS broadcast
                const float diag = A_lds[(rb + m) * APAD + rb + m];
                y[m] = (s + dt2 * accv) / (1.0f - dt2 * diag);
                myz[m] = dt2 * y[m];                                  // Z_k, transposed
            }
            // final solution rows for this block -> out (64 B contiguous per lane)
            float* o = out + (size_t)(batch_base + lane) * SN + rb;
            #pragma unroll
            for (int m = 0; m < 16; m += 4) {
                v4f t = { y[m], y[m + 1], y[m + 2], y[m + 3] };
                *(v4f*)(o + m) = t;
            }
        }
        asm volatile("s_wait_dscnt 0x0" ::: "memory");
    }
}

extern "C" void kernel_launch(void* const* d_in, const int* in_sizes, int n_in,
                              void* d_out, int out_size, void* d_ws, size_t ws_size,
                              hipStream_t stream) {
    const float* u     = (const float*)d_in[0];
    const float* v     = (const float*)d_in[1];
    const float* delta = (const float*)d_in[2];
    const float* A     = (const float*)d_in[3];
    const float* Bvec  = (const float*)d_in[4];
    float* out = (float*)d_out;

    const int batch  = in_sizes[0] / SN;                 // 4096
    const int blocks = batch / (WAVES * TILE);           // 128 workgroups of 2 waves
    hippo_bilinear_wmma<<<blocks, WAVES * 32, 0, stream>>>(u, v, delta, A, Bvec, out, batch);
}